// BayesianLSTM_49383533969599
// MI455X (gfx1250) — compile-verified
//
#include <hip/hip_runtime.h>
#include <math.h>

// ---------------------------------------------------------------------------
// Bayesian LSTM on MI455X (gfx1250), round 2.
// Roofline: HBM-bound on per-step noise Weps (538 MB total -> ~23us floor at
// 23.3 TB/s). f32 WMMA (16x16x4) since compute (~275 GFLOP) is negligible.
// This round: (a) W_t = Wmu + sigW.*Weps[t] sampled ONCE per step by a small
// float4 kernel (stays in L2), so the GEMM streams 1 tensor instead of 3;
// (b) each wave keeps a 64x32 accumulator tile so every B fragment in
// registers feeds 4 WMMAs -> inner loop is 8 loads : 8 wmma (was ~50 : 8);
// (c) gate tiles are exchanged through XOR-swizzled LDS for the fused
// sigmoid/tanh LSTM update (no gate tensor ever reaches global memory).
// ---------------------------------------------------------------------------

typedef __attribute__((ext_vector_type(2))) float v2f;
typedef __attribute__((ext_vector_type(8))) float v8f;

#define B_  512
#define T_  128
#define H_  512
#define I_  513      // 1 + H
#define M_  1024     // B * 2 combined rows

// ---------------------------------------------------------------------------
// sigma = softplus(rho) = log1p(exp(rho)); rho in [-5,-4], no overflow risk
// ---------------------------------------------------------------------------
__global__ void softplus_kernel(const float* __restrict__ src,
                                float* __restrict__ dst, int n) {
  int i = blockIdx.x * blockDim.x + threadIdx.x;
  if (i < n) dst[i] = log1pf(__expf(src[i]));
}

// ---------------------------------------------------------------------------
// Per-step weight sampling: Wt = Wmu + sigW .* WepsT   (float4 vectorized,
// n/4 threads). This is the only kernel that streams Weps[t] from HBM.
// ---------------------------------------------------------------------------
__global__ void sample_w_kernel(const float4* __restrict__ wmu,
                                const float4* __restrict__ sigW,
                                const float4* __restrict__ wepsT,
                                float4* __restrict__ Wt, int n4) {
  int i = blockIdx.x * blockDim.x + threadIdx.x;
  if (i >= n4) return;
  float4 m = wmu[i], s = sigW[i], e = wepsT[i], o;
  o.x = __builtin_fmaf(s.x, e.x, m.x);
  o.y = __builtin_fmaf(s.y, e.y, m.y);
  o.z = __builtin_fmaf(s.z, e.z, m.z);
  o.w = __builtin_fmaf(s.w, e.w, m.w);
  Wt[i] = o;
}

// ---------------------------------------------------------------------------
// One fused LSTM timestep.
// Grid (M_/128, H_/32) = (8,16), block 256 = 8 wave32.
// Wave w: gate g = w&3, m-half mh = w>>2 -> rows [mh*64, +64) of the 128-row
// tile, 32 columns. acc[4 m-subtiles][2 n-subtiles] of 16x16 f32.
// K-loop: per 4-K chunk, 2 B frags (4 loads) + 4 A frags (4 v2f) -> 8 wmma;
// each B frag is reused by 4 wmmas from registers.
// Epilogue: gate tiles -> LDS (XOR-swizzled, conflict-free), barrier, then
// each wave updates 16 rows x 32 cols: rank-1 x column + bias + LSTM update.
// ---------------------------------------------------------------------------
__global__ __launch_bounds__(256) void lstm_step_kernel(
    int t,
    const float* __restrict__ x,       // [B,T,2]
    const float* __restrict__ Hprev,   // [M_,H_]
    const float* __restrict__ Cprev,   // [M_,H_]
    float* __restrict__ Hnext,         // [M_,H_]
    float* __restrict__ Cnext,         // [M_,H_]
    const float* __restrict__ Wt,      // [4,I_,H_] sampled weights (this step)
    const float* __restrict__ bmu,     // [4,H_]
    const float* __restrict__ sigB,    // [4,H_]
    const float* __restrict__ bepsT)   // [4,H_]  (Beps + t*4*H_)
{
  __shared__ float lds[4 * 128 * 32];  // [gate][m_local][n^swz]  = 64 KB

  const int lane = threadIdx.x & 31;
  const int wave = threadIdx.x >> 5;
  const int g    = wave & 3;           // gate handled by this wave
  const int mh   = wave >> 2;          // m-half (0/1) of the 128-row tile
  const int l16  = lane & 15;
  const int kh   = lane >> 4;          // K-half select for A/B frags
  const int m_wg = blockIdx.x * 128;
  const int n_wg = blockIdx.y * 32;

  v8f acc[4][2] = {};

  // A 16x4 layout: lane<16 -> K = k,k+1 ; lane>=16 -> K = k+2,k+3
  const float* arow = Hprev + (size_t)(m_wg + mh * 64 + l16) * H_ + 2 * kh;
  // B 4x16 layout mirrors A; i = 1 + k + 2*kh + v  (i=0 x-column in epilogue)
  const float* brow = Wt + ((size_t)g * I_ + 1 + 2 * kh) * H_ + n_wg + l16;

  for (int k = 0; k < H_; k += 4) {
    const size_t bo = (size_t)k * H_;
    v2f bb0, bb1;
    bb0.x = brow[bo];        bb0.y = brow[bo + H_];        // ns = 0
    bb1.x = brow[bo + 16];   bb1.y = brow[bo + H_ + 16];   // ns = 1
#pragma unroll
    for (int mq = 0; mq < 4; ++mq) {
      v2f a;
      a.x = arow[(size_t)mq * 16 * H_ + k];
      a.y = arow[(size_t)mq * 16 * H_ + k + 1];
      acc[mq][0] = __builtin_amdgcn_wmma_f32_16x16x4_f32(
          false, a, false, bb0, (short)0, acc[mq][0], false, false);
      acc[mq][1] = __builtin_amdgcn_wmma_f32_16x16x4_f32(
          false, a, false, bb1, (short)0, acc[mq][1], false, false);
    }
  }

  // ---- stage gate tiles in LDS ------------------------------------------
  // C frag: M = v + 8*kh, N = l16. Column swizzle n ^ (((m>>3)&1)<<4);
  // here (m_local>>3)&1 == kh, so the store swizzle is just ^ (kh<<4).
  const int swz_st = kh << 4;
#pragma unroll
  for (int mq = 0; mq < 4; ++mq) {
#pragma unroll
    for (int ns = 0; ns < 2; ++ns) {
      const int ncol = ((16 * ns + l16) ^ swz_st);
#pragma unroll
      for (int v = 0; v < 8; ++v) {
        const int m_local = mh * 64 + mq * 16 + v + 8 * kh;
        lds[(g * 128 + m_local) * 32 + ncol] = acc[mq][ns][v];
      }
    }
  }
  __syncthreads();

  // ---- fused epilogue: rank-1 x column + bias + LSTM update --------------
  const int h = n_wg + lane;                     // one column per lane
  float wt0[4], bias[4];
#pragma unroll
  for (int gg = 0; gg < 4; ++gg) {
    wt0[gg]  = Wt[(size_t)gg * I_ * H_ + h];     // i = 0 row of W_t[gg]
    bias[gg] = __builtin_fmaf(sigB[gg * H_ + h], bepsT[gg * H_ + h],
                              bmu[gg * H_ + h]);
  }

#pragma unroll 4
  for (int r = 0; r < 16; ++r) {
    const int m_local = wave * 16 + r;           // wave handles 16 rows
    const int m = m_wg + m_local;
    const int swz = ((m_local >> 3) & 1) << 4;
    const int nc = lane ^ swz;
    const float xm = x[((size_t)(m >> 1) * T_ + t) * 2 + (m & 1)];
    const float g0 = lds[(0 * 128 + m_local) * 32 + nc] + xm * wt0[0] + bias[0];
    const float g1 = lds[(1 * 128 + m_local) * 32 + nc] + xm * wt0[1] + bias[1];
    const float g2 = lds[(2 * 128 + m_local) * 32 + nc] + xm * wt0[2] + bias[2];
    const float g3 = lds[(3 * 128 + m_local) * 32 + nc] + xm * wt0[3] + bias[3];
    const float ig = 1.0f / (1.0f + __expf(-g0));
    const float fg = 1.0f / (1.0f + __expf(-g1));
    const float ch = tanhf(g2);
    const size_t idx = (size_t)m * H_ + h;
    const float cn = fg * Cprev[idx] + ig * ch;
    Cnext[idx] = cn;
    Hnext[idx] = g3 * tanhf(cn);                 // note: no sigmoid on g3
  }
}

// ---------------------------------------------------------------------------
// out[b*2+c] = dot(H_T[b,c,:], fWmu + sp(fWrho).*fWeps) + fb   -> [B,1,2]
// ---------------------------------------------------------------------------
__global__ void final_proj_kernel(const float* __restrict__ Ht,
                                  const float* __restrict__ fwmu,
                                  const float* __restrict__ fwrho,
                                  const float* __restrict__ fbmu,
                                  const float* __restrict__ fbrho,
                                  const float* __restrict__ fweps,
                                  const float* __restrict__ fbeps,
                                  float* __restrict__ out) {
  const int m = blockIdx.x * blockDim.x + threadIdx.x;
  if (m >= M_) return;
  const float* hrow = Ht + (size_t)m * H_;
  float acc = 0.f;
  for (int hh = 0; hh < H_; ++hh) {
    const float fw = __builtin_fmaf(log1pf(__expf(fwrho[hh])), fweps[hh], fwmu[hh]);
    acc = __builtin_fmaf(hrow[hh], fw, acc);
  }
  out[m] = acc + __builtin_fmaf(log1pf(__expf(fbrho[0])), fbeps[0], fbmu[0]);
}

// ---------------------------------------------------------------------------
extern "C" void kernel_launch(void* const* d_in, const int* in_sizes, int n_in,
                              void* d_out, int out_size, void* d_ws, size_t ws_size,
                              hipStream_t stream) {
  const float* x     = (const float*)d_in[0];
  const float* H0    = (const float*)d_in[1];
  const float* C0    = (const float*)d_in[2];
  const float* Wmu   = (const float*)d_in[3];
  const float* Wrho  = (const float*)d_in[4];
  const float* Bmu   = (const float*)d_in[5];
  const float* Brho  = (const float*)d_in[6];
  const float* fWmu  = (const float*)d_in[7];
  const float* fWrho = (const float*)d_in[8];
  const float* fBmu  = (const float*)d_in[9];
  const float* fBrho = (const float*)d_in[10];
  const float* Weps  = (const float*)d_in[11];
  const float* Beps  = (const float*)d_in[12];
  const float* fWeps = (const float*)d_in[13];
  const float* fBeps = (const float*)d_in[14];
  float* out = (float*)d_out;

  // ws (floats): sigW[4*I*H] | sigB[4*H] | Wt[4*I*H] | H ping-pong | C ping-pong
  const size_t nW  = (size_t)4 * I_ * H_;   // 1,050,624 (x4B, 16B aligned)
  const size_t nHC = (size_t)M_ * H_;
  float* ws   = (float*)d_ws;
  float* sigW = ws;
  float* sigB = sigW + nW;
  float* Wt   = sigB + 4 * H_;
  float* Hb0  = Wt + nW;
  float* Hb1  = Hb0 + nHC;
  float* Cb0  = Hb1 + nHC;
  float* Cb1  = Cb0 + nHC;

  // 1) timestep-invariant softplus of rho parameters
  softplus_kernel<<<dim3((unsigned)((nW + 255) / 256)), 256, 0, stream>>>(Wrho, sigW, (int)nW);
  softplus_kernel<<<dim3((4 * H_ + 255) / 256), 256, 0, stream>>>(Brho, sigB, 4 * H_);

  // 2) seed recurrent state
  hipMemcpyAsync(Hb0, H0, sizeof(float) * nHC, hipMemcpyDeviceToDevice, stream);
  hipMemcpyAsync(Cb0, C0, sizeof(float) * nHC, hipMemcpyDeviceToDevice, stream);

  // 3) sequential scan: sample W_t, then fused WMMA step (ping-pong H/C)
  const int n4 = (int)(nW / 4);
  dim3 sgrid((n4 + 255) / 256);
  dim3 ggrid(M_ / 128, H_ / 32);   // 8 x 16 = 128 workgroups, 256 thr each
  for (int t = 0; t < T_; ++t) {
    float* Hp = (t & 1) ? Hb1 : Hb0;
    float* Cp = (t & 1) ? Cb1 : Cb0;
    float* Hn = (t & 1) ? Hb0 : Hb1;
    float* Cn = (t & 1) ? Cb0 : Cb1;
    sample_w_kernel<<<sgrid, 256, 0, stream>>>(
        (const float4*)Wmu, (const float4*)sigW,
        (const float4*)(Weps + (size_t)t * nW), (float4*)Wt, n4);
    lstm_step_kernel<<<ggrid, 256, 0, stream>>>(
        t, x, Hp, Cp, Hn, Cn, Wt, Bmu, sigB, Beps + (size_t)t * 4 * H_);
  }

  // 4) final projection; T_ even so last H landed in Hb0
  final_proj_kernel<<<(M_ + 127) / 128, 128, 0, stream>>>(
      Hb0, fWmu, fWrho, fBmu, fBrho, fWeps, fBeps, out);
}